// RPINetEncoder_32590211842558
// MI455X (gfx1250) — compile-verified
//
#include <hip/hip_runtime.h>

typedef __attribute__((ext_vector_type(2))) float v2f;
typedef __attribute__((ext_vector_type(8))) float v8f;

#define N_NODES 100000
#define N_EDGES 800000
#define DIM 64
#define M_TILES (N_NODES / 16)   // 6250, exact
#define PAIR_PITCH 80            // float2 pitch; 80*2 dwords == 32 mod 64 ->
                                 // lanes 0-15 / 16-31 hit complementary banks

// ---------------------------------------------------------------------------
// Zero the aggregation buffer (graph-capture safe; no hipMemsetAsync needed).
// ---------------------------------------------------------------------------
__global__ __launch_bounds__(256) void zero_kernel(float4* __restrict__ p, int n4) {
    int i = blockIdx.x * 256 + threadIdx.x;
    int stride = gridDim.x * 256;
    float4 z = make_float4(0.f, 0.f, 0.f, 0.f);
    for (; i < n4; i += stride) p[i] = z;
}

// ---------------------------------------------------------------------------
// Scatter: 2 edges per wave32. Lanes 0-15 handle edge A, lanes 16-31 edge B;
// each lane owns a float4 chunk of the 64-wide row, so each half-wave issues
// one coalesced 256B global_load_b128 row read, then 4 native f32 global
// atomic adds (no return -> STOREcnt path). agg is L2-resident (25.6 MB).
// ---------------------------------------------------------------------------
__global__ __launch_bounds__(256) void scatter_kernel(const float* __restrict__ x,
                                                      const long long* __restrict__ edge_index,
                                                      const float* __restrict__ edge_weight,
                                                      float* __restrict__ agg,
                                                      int n_edges) {
    int g = blockIdx.x * 256 + threadIdx.x;
    int wave = g >> 5;
    int lane = g & 31;
    int e = wave * 2 + (lane >> 4);     // half-wave picks its edge
    if (e >= n_edges) return;
    int l16 = lane & 15;

    long long src = edge_index[e];            // row 0 of (2, E)
    long long dst = edge_index[n_edges + e];  // row 1 of (2, E)
    float w = edge_weight[e];

    const float4 xv = *(const float4*)(x + src * DIM + l16 * 4);
    float* dp = agg + dst * DIM + l16 * 4;
    unsafeAtomicAdd(dp + 0, w * xv.x);   // -> global_atomic_add_f32 (no rtn)
    unsafeAtomicAdd(dp + 1, w * xv.y);
    unsafeAtomicAdd(dp + 2, w * xv.z);
    unsafeAtomicAdd(dp + 3, w * xv.w);
}

// ---------------------------------------------------------------------------
// out = relu(agg @ W^T + b) via V_WMMA_F32_16X16X4_F32.
// One wave per 16-row tile; 4 n-tiles of 16 cols; K-loop 64 in steps of 4.
//
// A (16x4 f32): lane<16 -> K={k0,k0+1}, lane>=16 -> K={k0+2,k0+3}, M=lane&15.
// B (4x16):     same K split, N=lane&15.  B[k][n] = W[n][k].
// B is staged in LDS as float2 pairs sB[k/2][n] = {WT[k][n], WT[k+1][n]} so a
// lane's whole fragment is ONE ds_load_b64 into an adjacent even VGPR pair —
// no repacking movs before the WMMA.
// C/D (16x16):  vgpr v: M = v + (lane<16 ? 0 : 8), N = lane&15.
// ---------------------------------------------------------------------------
__global__ __launch_bounds__(128) void gemm_relu_kernel(const float* __restrict__ agg,
                                                        const float* __restrict__ W,
                                                        const float* __restrict__ bias,
                                                        float* __restrict__ out,
                                                        int n_tiles) {
    __shared__ float2 sB[32 * PAIR_PITCH];   // 20 KB

    // Cooperative build: W[n][k] -> sB[k>>1][n] component (k&1).
    {
        float* sBf = (float*)sB;
        for (int i = threadIdx.x; i < DIM * DIM; i += blockDim.x) {
            int n = i >> 6;      // row of W
            int k = i & 63;      // col of W
            sBf[((k >> 1) * PAIR_PITCH + n) * 2 + (k & 1)] = W[i];
        }
    }
    __syncthreads();

    int wave = threadIdx.x >> 5;
    int lane = threadIdx.x & 31;
    int tile = blockIdx.x * 4 + wave;
    if (tile >= n_tiles) return;

    const int half  = (lane < 16) ? 0 : 2;    // K sub-offset
    const int phalf = (lane < 16) ? 0 : 1;    // pair-row sub-offset
    const int m_a   = tile * 16 + (lane & 15);
    const int n_b   = lane & 15;

    v8f c0 = {}, c1 = {}, c2 = {}, c3 = {};

#pragma unroll
    for (int k0 = 0; k0 < DIM; k0 += 4) {
        // A fragment: one global b64 at agg[m_a][k0+half].
        const float2 av = *(const float2*)(agg + m_a * DIM + k0 + half);
        v2f a; a.x = av.x; a.y = av.y;

        const int p = (k0 >> 1) + phalf;      // pair row in sB
        const float2* prow = sB + p * PAIR_PITCH + n_b;

        float2 f0 = prow[0];
        float2 f1 = prow[16];
        float2 f2 = prow[32];
        float2 f3 = prow[48];
        v2f b0; b0.x = f0.x; b0.y = f0.y;
        v2f b1; b1.x = f1.x; b1.y = f1.y;
        v2f b2; b2.x = f2.x; b2.y = f2.y;
        v2f b3; b3.x = f3.x; b3.y = f3.y;

        c0 = __builtin_amdgcn_wmma_f32_16x16x4_f32(false, a, false, b0, (short)0, c0, false, false);
        c1 = __builtin_amdgcn_wmma_f32_16x16x4_f32(false, a, false, b1, (short)0, c1, false, false);
        c2 = __builtin_amdgcn_wmma_f32_16x16x4_f32(false, a, false, b2, (short)0, c2, false, false);
        c3 = __builtin_amdgcn_wmma_f32_16x16x4_f32(false, a, false, b3, (short)0, c3, false, false);
    }

    // Epilogue: bias + ReLU + store.
    const int rbase = tile * 16 + ((lane < 16) ? 0 : 8);
    const int cbase = lane & 15;
    const float bb0 = bias[ 0 + cbase];
    const float bb1 = bias[16 + cbase];
    const float bb2 = bias[32 + cbase];
    const float bb3 = bias[48 + cbase];

#pragma unroll
    for (int v = 0; v < 8; ++v) {
        float* row = out + (rbase + v) * DIM;
        float r0 = c0[v] + bb0; row[ 0 + cbase] = r0 > 0.f ? r0 : 0.f;
        float r1 = c1[v] + bb1; row[16 + cbase] = r1 > 0.f ? r1 : 0.f;
        float r2 = c2[v] + bb2; row[32 + cbase] = r2 > 0.f ? r2 : 0.f;
        float r3 = c3[v] + bb3; row[48 + cbase] = r3 > 0.f ? r3 : 0.f;
    }
}

// ---------------------------------------------------------------------------
// Driver: 3 iterations of (zero agg -> scatter -> wmma gemm+relu), ping-pong
// between d_out and a workspace buffer so the final iteration lands in d_out.
// ---------------------------------------------------------------------------
extern "C" void kernel_launch(void* const* d_in, const int* in_sizes, int n_in,
                              void* d_out, int out_size, void* d_ws, size_t ws_size,
                              hipStream_t stream) {
    const float*     x  = (const float*)d_in[0];
    const long long* ei = (const long long*)d_in[1];   // int64 (2, E)
    const float*     ew = (const float*)d_in[2];
    const float*     W  = (const float*)d_in[3];
    const float*     b  = (const float*)d_in[4];
    float* out = (float*)d_out;

    float* agg  = (float*)d_ws;                                        // 25.6 MB
    float* xbuf = (float*)((char*)d_ws + (size_t)N_NODES * DIM * 4);   // 25.6 MB

    const int n_elems   = N_NODES * DIM;
    const int zero_grid = 2048;                                  // grid-stride
    const int n_waves   = (N_EDGES + 1) / 2;                     // 2 edges/wave
    const int scat_grid = (n_waves * 32 + 255) / 256;
    const int gemm_grid = (M_TILES + 3) / 4;                     // 4 tiles/block

    const float* cur = x;
    for (int it = 0; it < 3; ++it) {
        float* nxt = (it == 1) ? xbuf : out;
        zero_kernel<<<zero_grid, 256, 0, stream>>>((float4*)agg, n_elems / 4);
        scatter_kernel<<<scat_grid, 256, 0, stream>>>(cur, ei, ew, agg, N_EDGES);
        gemm_relu_kernel<<<gemm_grid, 128, 0, stream>>>(agg, W, b, nxt, M_TILES);
        cur = nxt;
    }
}